// Time_Pos_Encoding_47124381171737
// MI455X (gfx1250) — compile-verified
//
#include <hip/hip_runtime.h>
#include <math.h>

// out[b,s,d] = x[b,s,d] + pe[s,d]
//   div      = 10000^(2*i/D)          (raw index i, per reference)
//   pe[s,i]  = i even ? sin(s/div)+sin(T/div) : cos(s/div)+cos(T/div)
// x: [8, 4096, 1024] fp32. Memory-bound: ~268 MB HBM floor (~11.5us @ 23.3TB/s).

typedef float v4f __attribute__((ext_vector_type(4)));

#define PE_S 4096
#define PE_D 1024
#define PE_TIME 101.0f
#define LOG2_10000 13.2877123795494f   // log2(10000)

// pe values for 4 consecutive feature indices [dbase, dbase+3] at row s
__device__ __forceinline__ v4f pe_quad(int s, int dbase) {
    v4f r;
    const float sf = (float)s;
    #pragma unroll
    for (int j = 0; j < 4; ++j) {
        const int i = dbase + j;
        // inv_div = 10000^(-2*i/D) = exp2(-2*i/D * log2(10000))
        const float inv_div = exp2f(-2.0f * (float)i * (1.0f / (float)PE_D) * LOG2_10000);
        const float ang_pos = sf * inv_div;
        const float ang_t   = PE_TIME * inv_div;
        float v;
        if ((i & 1) == 0) v = sinf(ang_pos) + sinf(ang_t);
        else              v = cosf(ang_pos) + cosf(ang_t);
        r[j] = v;
    }
    return r;
}

// ---- Kernel 1: materialize pe[S*D] into workspace (runs once per launch, tiny) ----
__global__ void pe_init_kernel(float* __restrict__ pe) {
    const int idx4 = blockIdx.x * blockDim.x + threadIdx.x;   // one float4 per lane
    if (idx4 >= (PE_S * PE_D) / 4) return;
    const int s     = idx4 >> 8;             // D/4 = 256 quads per row
    const int dbase = (idx4 & 255) << 2;
    // Regular-temporal store: we WANT pe resident in the 192MB L2 for reuse.
    *(v4f*)(pe + idx4 * 4) = pe_quad(s, dbase);
}

// ---- Kernel 2: streaming add. x/out non-temporal (don't pollute L2); pe RT (stays in L2) ----
__global__ void pe_add_kernel(const float* __restrict__ x,
                              const float* __restrict__ pe,
                              float* __restrict__ out, int n4) {
    const int idx = blockIdx.x * blockDim.x + threadIdx.x;
    if (idx >= n4) return;
    const int pidx = idx & ((PE_S * PE_D) / 4 - 1);           // broadcast over batch (pow2 mask)
    const v4f xv = __builtin_nontemporal_load((const v4f*)(x + (size_t)idx * 4));
    const v4f pv = *(const v4f*)(pe + (size_t)pidx * 4);      // RT load -> L2-resident
    v4f o;
    o[0] = xv[0] + pv[0]; o[1] = xv[1] + pv[1];
    o[2] = xv[2] + pv[2]; o[3] = xv[3] + pv[3];
    __builtin_nontemporal_store(o, (v4f*)(out + (size_t)idx * 4));
}

// ---- Fallback: fused (pe computed inline) if workspace is too small ----
__global__ void pe_fused_kernel(const float* __restrict__ x,
                                float* __restrict__ out, int n4) {
    const int idx = blockIdx.x * blockDim.x + threadIdx.x;
    if (idx >= n4) return;
    const int sd4   = idx & ((PE_S * PE_D) / 4 - 1);
    const int s     = sd4 >> 8;
    const int dbase = (sd4 & 255) << 2;
    const v4f xv = __builtin_nontemporal_load((const v4f*)(x + (size_t)idx * 4));
    const v4f pv = pe_quad(s, dbase);
    v4f o;
    o[0] = xv[0] + pv[0]; o[1] = xv[1] + pv[1];
    o[2] = xv[2] + pv[2]; o[3] = xv[3] + pv[3];
    __builtin_nontemporal_store(o, (v4f*)(out + (size_t)idx * 4));
}

extern "C" void kernel_launch(void* const* d_in, const int* in_sizes, int n_in,
                              void* d_out, int out_size, void* d_ws, size_t ws_size,
                              hipStream_t stream) {
    const float* x = (const float*)d_in[0];
    float* out = (float*)d_out;

    const int n  = in_sizes[0];          // 8*4096*1024 = 33,554,432
    const int n4 = n >> 2;               // float4 count
    const int threads = 256;             // 8 wave32s

    const size_t pe_bytes = (size_t)PE_S * PE_D * sizeof(float);   // 16 MiB

    if (d_ws != nullptr && ws_size >= pe_bytes) {
        float* pe = (float*)d_ws;
        const int pe_quads = (PE_S * PE_D) / 4;                    // 1,048,576
        pe_init_kernel<<<(pe_quads + threads - 1) / threads, threads, 0, stream>>>(pe);
        pe_add_kernel<<<(n4 + threads - 1) / threads, threads, 0, stream>>>(x, pe, out, n4);
    } else {
        pe_fused_kernel<<<(n4 + threads - 1) / threads, threads, 0, stream>>>(x, out, n4);
    }
}